// DynamicOptimizerModule_16295105921343
// MI455X (gfx1250) — compile-verified
//
#include <hip/hip_runtime.h>

// ---------------------------------------------------------------------------
// out[128,16384] = A[128,128] @ B[128,16384]   (f32 GEMM via V_WMMA_F32_16X16X4_F32)
//   A[m][k]  = sum of weights[e] where dst[e]==m+128 and src[e]==k
//   B row r  = loss (r==0), prev_loss (r==1), params[r-2] (r>=2)
//
// Roofline: 16 MB HBM traffic (read B 8MB once + write C 8MB) ~= 0.7us @
// 23.3 TB/s; 0.54 GFLOP of f32 WMMA hides under it. A-build is one pass of
// deterministic integer fixed-point atomics (associative => order-invariant).
// M is split 4-ways (grid 128x4 = 512 blocks = 4096 waves) for latency hiding;
// the extra B re-reads are served by the 192MB L2 (B is only 8MB).
// ---------------------------------------------------------------------------

typedef __attribute__((ext_vector_type(2))) float v2f;
typedef __attribute__((ext_vector_type(8))) float v8f;

#define D_DIM 16384
#define S_DIM 128                 // K (2 + 126 feature rows)
#define NOUT  128                 // M (nodes 128..255)
#define E_DIM 4096
#define MT_PER_WAVE 2             // 16x16 accum tiles per wave
#define M_PER_BLOCK (MT_PER_WAVE * 16)  // 32 rows per block
#define LDS_PITCH 132             // 128+4: conflict-free ds_load_b64 of A frags
#define FP_SCALE 4294967296.0f    // 2^32 fixed-point scale for deterministic adds
#define FP_INV   (1.0f / 4294967296.0f)

// ------------------------- A build (deterministic) -------------------------
__global__ __launch_bounds__(256) void zero_kernel(unsigned long long* __restrict__ Ai) {
  const int i = blockIdx.x * 256 + threadIdx.x;
  if (i < NOUT * S_DIM) Ai[i] = 0ull;
}

// One pass over edges. Integer (two's-complement) atomicAdd is associative and
// commutative => bit-identical result for any arrival order, unlike f32 atomics.
__global__ __launch_bounds__(256) void scatter_kernel(
    const float* __restrict__ w, const int* __restrict__ src,
    const int* __restrict__ dst, unsigned long long* __restrict__ Ai) {
  const int e = blockIdx.x * 256 + threadIdx.x;
  if (e >= E_DIM) return;
  const int d = dst[e];
  if (d < S_DIM) return;  // those node slots are overwritten by features; not output
  const long long q = (long long)llrintf(w[e] * FP_SCALE);
  atomicAdd(&Ai[(size_t)(d - S_DIM) * S_DIM + src[e]], (unsigned long long)q);
}

// ------------------------------- GEMM --------------------------------------
__device__ __forceinline__ const float* feat_row(
    const float* __restrict__ loss, const float* __restrict__ prev,
    const float* __restrict__ params, int r) {
  const float* p = params + (size_t)(r >= 2 ? r - 2 : 0) * D_DIM;
  p = (r == 1) ? prev : p;
  p = (r == 0) ? loss : p;
  return p;
}

__global__ __launch_bounds__(256) void gemm_kernel(
    const float* __restrict__ loss, const float* __restrict__ prev,
    const float* __restrict__ params, const long long* __restrict__ Ai,
    float* __restrict__ out) {
  // 32 rows x 132-float pitch = 16.9KB of the 320KB WGP LDS.
  __shared__ __align__(16) float As[M_PER_BLOCK * LDS_PITCH];

  const int tid   = threadIdx.x;
  const int mbase = blockIdx.y * M_PER_BLOCK;

  // Cooperative stage of this block's A slice, converting fixed-point -> f32.
  for (int i = tid; i < M_PER_BLOCK * S_DIM; i += 256) {
    const int m = i >> 7;      // /128
    const int k = i & 127;
    As[m * LDS_PITCH + k] = (float)Ai[(size_t)(mbase + m) * S_DIM + k] * FP_INV;
  }
  __syncthreads();

  const int wave = tid >> 5;        // 0..7 : 16-column slice within block
  const int lane = tid & 31;
  const int half = lane >> 4;       // 0 or 1
  const int l16  = lane & 15;
  const int n    = blockIdx.x * 128 + wave * 16 + l16;  // global column

  v8f c[MT_PER_WAVE];
#pragma unroll
  for (int mt = 0; mt < MT_PER_WAVE; ++mt) c[mt] = (v8f)(0.f);

  // K loop: 32 steps of K=4.
  for (int kk = 0; kk < 32; ++kk) {
    const int k0 = kk * 4;
    // B fragment (4x16): lane = column n; lanes 0-15 rows {k0,k0+1},
    // lanes 16-31 rows {k0+2,k0+3}  (ISA 32-bit operand layout).
    const int rb = k0 + 2 * half;
    v2f b;
    b.x = feat_row(loss, prev, params, rb)[n];
    b.y = feat_row(loss, prev, params, rb + 1)[n];

#pragma unroll
    for (int mt = 0; mt < MT_PER_WAVE; ++mt) {
      // A fragment (16x4): lane row = l16; lanes 0-15 cols {k0,k0+1},
      // lanes 16-31 cols {k0+2,k0+3}.
      const int m = mt * 16 + l16;
      const v2f a = *(const v2f*)(As + m * LDS_PITCH + k0 + 2 * half);
      c[mt] = __builtin_amdgcn_wmma_f32_16x16x4_f32(
          false, a, false, b, (short)0, c[mt], false, false);
    }
  }

  // C/D 16x16 f32 layout: VGPR v = row v (lanes 0-15) / v+8 (lanes 16-31).
#pragma unroll
  for (int mt = 0; mt < MT_PER_WAVE; ++mt) {
#pragma unroll
    for (int v = 0; v < 8; ++v) {
      const int m = mbase + mt * 16 + v + 8 * half;
      out[(size_t)m * D_DIM + n] = c[mt][v];
    }
  }
}

// ------------------------------ launcher -----------------------------------
extern "C" void kernel_launch(void* const* d_in, const int* in_sizes, int n_in,
                              void* d_out, int out_size, void* d_ws, size_t ws_size,
                              hipStream_t stream) {
  const float* loss    = (const float*)d_in[0];
  const float* prev    = (const float*)d_in[1];
  const float* params  = (const float*)d_in[2];
  const float* weights = (const float*)d_in[3];
  const int*   src     = (const int*)d_in[4];
  const int*   dst     = (const int*)d_in[5];
  float* out = (float*)d_out;
  unsigned long long* Ai = (unsigned long long*)d_ws;  // 128*128 u64 = 128KB scratch

  zero_kernel<<<(NOUT * S_DIM + 255) / 256, 256, 0, stream>>>(Ai);
  scatter_kernel<<<(E_DIM + 255) / 256, 256, 0, stream>>>(weights, src, dst, Ai);
  gemm_kernel<<<dim3(D_DIM / 128, NOUT / M_PER_BLOCK), 256, 0, stream>>>(
      loss, prev, params, (const long long*)Ai, out);
}